// IVT_Encoder_19628000543289
// MI455X (gfx1250) — compile-verified
//
#include <hip/hip_runtime.h>
#include <hip/hip_bf16.h>
#include <math.h>

// ---------------------------------------------------------------------------
// The whole reference collapses: layernorm over a singleton dim makes the
// network input a constant (norm_b), so the Mamba scans run on a constant
// sequence (one wave32 per direction, 32 (d,n) states = 32 lanes), the MLP is
// three mat-vecs computed once via V_WMMA_F32_16X16X4_F32, and the output is
// a 100-float vector broadcast to 8192 rows (3.28 MB store, HBM-bound).
// ---------------------------------------------------------------------------

typedef float v2f __attribute__((ext_vector_type(2)));
typedef float v8f __attribute__((ext_vector_type(8)));

struct MambaParams {
    const float* norm_b;      // (1,)
    const float* in_proj_w;   // (4,)
    const float* out_proj_w;  // (2,)
    const float* conv_w[3];   // (2,4)
    const float* conv_b[3];   // (2,)
    const float* xproj_w[3];  // (33,2)
    const float* dtproj_w[3]; // (2,)
    const float* dtproj_b[3]; // (2,)
    const float* A_log[3];    // (2,16)
    const float* Dskip[3];    // (2,)
    float*       y_out;       // ws[0..255]
};

__device__ __forceinline__ float silu_f(float v)  { return v / (1.0f + __expf(-v)); }
__device__ __forceinline__ float sftp_f(float v)  { return (v > 20.0f) ? v : log1pf(__expf(v)); }
__device__ __forceinline__ float elu_f(float v)   { return v > 0.0f ? v : (__expf(v) - 1.0f); }

// ---------------------------------------------------------------------------
// Kernel 1: three Mamba directions (constant input sequence) + combine.
// Wave w < 3 owns direction w; lane = d*16 + n owns SSM state (d, n).
// Only 4 distinct step types exist (conv left-padding l=0,1,2, then l>=3).
// ---------------------------------------------------------------------------
__global__ __launch_bounds__(128) void mamba_scan_kernel(MambaParams P) {
    __shared__ float ydir[3][256];
    const int tid  = threadIdx.x;
    const int wave = tid >> 5;
    const int lane = tid & 31;

    if (wave < 3) {
        const int dir = wave;
        const int d = lane >> 4;
        const int n = lane & 15;

        const float nb = P.norm_b[0];
        float cx[2], gz[2];
        #pragma unroll
        for (int dd = 0; dd < 2; ++dd) {
            cx[dd] = nb * P.in_proj_w[dd];
            gz[dd] = silu_f(nb * P.in_proj_w[2 + dd]);
        }

        const float* cw  = P.conv_w[dir];
        const float* cb  = P.conv_b[dir];
        const float* xp  = P.xproj_w[dir];
        const float* dtw = P.dtproj_w[dir];
        const float* dtb = P.dtproj_b[dir];
        const float* Al  = P.A_log[dir];
        const float* Dk  = P.Dskip[dir];

        // u[t][dd] = silu(conv)(position class t), t = min(l,3)
        float u[4][2];
        #pragma unroll
        for (int dd = 0; dd < 2; ++dd) {
            float wsum = 0.0f;
            #pragma unroll
            for (int t = 0; t < 4; ++t) {
                wsum += cw[dd * 4 + (3 - t)];
                u[t][dd] = silu_f(cx[dd] * wsum + cb[dd]);
            }
        }

        const float Aval = -__expf(Al[d * 16 + n]);
        float decay[4], inject[4], Cv[4], skipadd[4];
        #pragma unroll
        for (int t = 0; t < 4; ++t) {
            const float x0 = xp[0] * u[t][0] + xp[1] * u[t][1];           // xdbl row 0
            const float dl = sftp_f(x0 * dtw[d] + dtb[d]);                // delta_d
            const float Bv = xp[(1  + n) * 2 + 0] * u[t][0] + xp[(1  + n) * 2 + 1] * u[t][1];
            Cv[t]          = xp[(17 + n) * 2 + 0] * u[t][0] + xp[(17 + n) * 2 + 1] * u[t][1];
            decay[t]   = __expf(dl * Aval);
            inject[t]  = dl * u[t][d] * Bv;
            skipadd[t] = u[t][d] * Dk[d];
        }

        const float gd = gz[d];
        const float wo = P.out_proj_w[d];

        float h = 0.0f;
        for (int l = 0; l < 256; ++l) {
            const int t = l < 3 ? l : 3;
            h = fmaf(decay[t], h, inject[t]);
            float v = h * Cv[t];
            // reduce over n (16 lanes within each half)
            v += __shfl_xor(v, 1);
            v += __shfl_xor(v, 2);
            v += __shfl_xor(v, 4);
            v += __shfl_xor(v, 8);
            float val = (v + skipadd[t]) * gd * wo;   // per-d scale, then sum over d
            val += __shfl_xor(val, 16);
            if (lane == 0) ydir[dir][l] = val;
        }
    }
    __syncthreads();

    // combine: forward + flipped backward + de-sliced slice direction
    for (int l = tid; l < 256; l += 128) {
        const float yv = ydir[0][l]
                       + ydir[1][255 - l]
                       + ydir[2][(l & 3) * 64 + (l >> 2)];
        P.y_out[l] = yv;
    }
}

// ---------------------------------------------------------------------------
// WMMA mat-vec tile: 16 output rows of  out = W(rows x K) * x  per wave.
// A: W rows (lane<16: K0/K1, lane>=16: K2/K3 per ISA 16x4 f32 layout).
// B: x chunk broadcast across all 16 columns -> D replicated over N.
// ---------------------------------------------------------------------------
__device__ __forceinline__ v8f wmma_matvec_tile(const float* __restrict__ W, int ldw,
                                                int row0, int nrows_valid,
                                                const float* __restrict__ xvec, int K,
                                                int lane) {
    v8f acc = {};
    const int  mrow = row0 + (lane & 15);
    const int  ksel = (lane >> 4) ? 2 : 0;
    const bool rv   = mrow < nrows_valid;
    for (int k0 = 0; k0 < K; k0 += 4) {
        const int kb = k0 + ksel;
        v2f a, b;
        if (rv) { a.x = W[mrow * ldw + kb]; a.y = W[mrow * ldw + kb + 1]; }
        else    { a.x = 0.0f;               a.y = 0.0f; }
        b.x = xvec[kb];
        b.y = xvec[kb + 1];
        acc = __builtin_amdgcn_wmma_f32_16x16x4_f32(false, a, false, b,
                                                    (short)0, acc, false, false);
    }
    return acc;
}

// ---------------------------------------------------------------------------
// Kernel 2: 256 -> 512 -> 256 -> 100 MLP, computed once (rows identical).
// 16 waves; each wave produces a 16-row output tile via chained WMMAs.
// ---------------------------------------------------------------------------
__global__ __launch_bounds__(512) void mlp_kernel(const float* __restrict__ w1, const float* __restrict__ b1,
                                                  const float* __restrict__ w2, const float* __restrict__ b2,
                                                  const float* __restrict__ w3, const float* __restrict__ b3,
                                                  const float* __restrict__ y_in,
                                                  float* __restrict__ out_vec) {
    __shared__ float yv[256];
    __shared__ float h1[512];
    __shared__ float h2[256];
    const int tid  = threadIdx.x;
    const int wave = tid >> 5;
    const int lane = tid & 31;

    if (tid < 256) yv[tid] = y_in[tid];
    __syncthreads();

    // layer 1: 512 outputs = 32 tiles over 16 waves
    for (int tile = wave; tile < 32; tile += 16) {
        const int r0 = tile * 16;
        v8f acc = wmma_matvec_tile(w1, 256, r0, 512, yv, 256, lane);
        if (lane == 0) {
            #pragma unroll
            for (int j = 0; j < 8; ++j) h1[r0 + j]     = elu_f(acc[j] + b1[r0 + j]);
        }
        if (lane == 16) {
            #pragma unroll
            for (int j = 0; j < 8; ++j) h1[r0 + 8 + j] = elu_f(acc[j] + b1[r0 + 8 + j]);
        }
    }
    __syncthreads();

    // layer 2: 256 outputs = 16 tiles
    {
        const int r0 = wave * 16;
        v8f acc = wmma_matvec_tile(w2, 512, r0, 256, h1, 512, lane);
        if (lane == 0) {
            #pragma unroll
            for (int j = 0; j < 8; ++j) h2[r0 + j]     = elu_f(acc[j] + b2[r0 + j]);
        }
        if (lane == 16) {
            #pragma unroll
            for (int j = 0; j < 8; ++j) h2[r0 + 8 + j] = elu_f(acc[j] + b2[r0 + 8 + j]);
        }
    }
    __syncthreads();

    // layer 3: 100 outputs = 7 tiles (rows >= 100 masked)
    if (wave < 7) {
        const int r0 = wave * 16;
        v8f acc = wmma_matvec_tile(w3, 256, r0, 100, h2, 256, lane);
        if (lane == 0) {
            #pragma unroll
            for (int j = 0; j < 8; ++j) {
                const int r = r0 + j;
                if (r < 100) out_vec[r] = acc[j] + b3[r];
            }
        }
        if (lane == 16) {
            #pragma unroll
            for (int j = 0; j < 8; ++j) {
                const int r = r0 + 8 + j;
                if (r < 100) out_vec[r] = acc[j] + b3[r];
            }
        }
    }
}

// ---------------------------------------------------------------------------
// Kernel 3: broadcast the 100-vector to all 8192 rows (pure store bandwidth).
// ---------------------------------------------------------------------------
__global__ __launch_bounds__(256) void broadcast_kernel(const float* __restrict__ vec,
                                                        float* __restrict__ out, int total) {
    const int idx = blockIdx.x * blockDim.x + threadIdx.x;
    if (idx < total) out[idx] = vec[idx % 100];
}

extern "C" void kernel_launch(void* const* d_in, const int* in_sizes, int n_in,
                              void* d_out, int out_size, void* d_ws, size_t ws_size,
                              hipStream_t stream) {
    (void)in_sizes; (void)n_in; (void)ws_size;
    float* ws = (float*)d_ws;     // ws[0..255] = y vector, ws[256..355] = out100

    MambaParams P;
    P.norm_b     = (const float*)d_in[2];
    P.in_proj_w  = (const float*)d_in[3];
    P.out_proj_w = (const float*)d_in[25];
    for (int dir = 0; dir < 3; ++dir) {
        const int base = 4 + 7 * dir;
        P.conv_w[dir]   = (const float*)d_in[base + 0];
        P.conv_b[dir]   = (const float*)d_in[base + 1];
        P.xproj_w[dir]  = (const float*)d_in[base + 2];
        P.dtproj_w[dir] = (const float*)d_in[base + 3];
        P.dtproj_b[dir] = (const float*)d_in[base + 4];
        P.A_log[dir]    = (const float*)d_in[base + 5];
        P.Dskip[dir]    = (const float*)d_in[base + 6];
    }
    P.y_out = ws;

    mamba_scan_kernel<<<1, 128, 0, stream>>>(P);

    mlp_kernel<<<1, 512, 0, stream>>>((const float*)d_in[26], (const float*)d_in[27],
                                      (const float*)d_in[28], (const float*)d_in[29],
                                      (const float*)d_in[30], (const float*)d_in[31],
                                      ws, ws + 256);

    const int total   = out_size;           // 8192 * 100
    const int threads = 256;
    const int blocks  = (total + threads - 1) / threads;
    broadcast_kernel<<<blocks, threads, 0, stream>>>(ws + 256, (float*)d_out, total);
}